// GATv2Conv_56908316672599
// MI455X (gfx1250) — compile-verified
//
#include <hip/hip_runtime.h>
#include <hip/hip_bf16.h>

typedef __attribute__((ext_vector_type(2))) float v2f;
typedef __attribute__((ext_vector_type(8))) float v8f;

#define NN   50000
#define EE   800000
#define DIN  128
#define HC   128   // H*C
#define HH   4
#define EDIM 64

__device__ __forceinline__ void atomicMaxF(float* addr, float v) {
    // signed-int ordering valid for v>=0, reversed unsigned ordering for v<0
    if (v >= 0.0f) atomicMax((int*)addr, __float_as_int(v));
    else           atomicMin((unsigned int*)addr, (unsigned int)__float_as_int(v));
}

// ---------------------------------------------------------------------------
// x_l = x @ W_l, x_r = x @ W_r  via V_WMMA_F32_16X16X4_F32.
// One wave per 16x16 output tile; blockIdx.z selects which weight/output.
// ---------------------------------------------------------------------------
__global__ void __launch_bounds__(32) node_gemm(const float* __restrict__ x,
                                                const float* __restrict__ Wl,
                                                const float* __restrict__ Wr,
                                                float* __restrict__ xl,
                                                float* __restrict__ xr) {
    const int lane = threadIdx.x;
    const int ln   = lane & 15;
    const int half = lane >> 4;
    const int tm   = blockIdx.x;           // 16-row tile
    const int tn   = blockIdx.y;           // 16-col tile
    const float* __restrict__ W   = blockIdx.z ? Wr : Wl;
    float* __restrict__       out = blockIdx.z ? xr : xl;

    // A frag: lanes 0-15 row = tm*16+ln holding k,k+1 ; lanes 16-31 same rows k+2,k+3
    const float* arow = x + (size_t)(tm * 16 + ln) * DIN + half * 2;
    // B frag: lanes 0-15 col = tn*16+ln rows k,k+1 ; lanes 16-31 rows k+2,k+3
    const float* bcol = W + tn * 16 + ln;

    v8f c = {};
#pragma unroll
    for (int kk = 0; kk < DIN / 4; ++kk) {
        v2f a, b;
        a.x = arow[kk * 4];
        a.y = arow[kk * 4 + 1];
        b.x = bcol[(size_t)(kk * 4 + half * 2) * HC];
        b.y = bcol[(size_t)(kk * 4 + half * 2 + 1) * HC];
        c = __builtin_amdgcn_wmma_f32_16x16x4_f32(false, a, false, b,
                                                  (short)0, c, false, false);
    }
    // D layout: VGPR v -> row v (lanes 0-15) / row v+8 (lanes 16-31)
    float* orow = out + (size_t)(tm * 16 + half * 8) * HC + tn * 16 + ln;
#pragma unroll
    for (int v = 0; v < 8; ++v) orow[(size_t)v * HC] = c[v];
}

// ---------------------------------------------------------------------------
// Fused: e_feat tile GEMM (WMMA f32) + gather x_l[src]/x_r[dst] + LeakyReLU +
// att dot product -> logits[E,H], plus segment atomic max into amax[N,H].
// One wave handles 16 edges (all 128 output channels).
// ---------------------------------------------------------------------------
__global__ void __launch_bounds__(32) edge_logits(const float* __restrict__ ea,
                                                  const float* __restrict__ We,
                                                  const float* __restrict__ att,
                                                  const int*   __restrict__ srcI,
                                                  const int*   __restrict__ dstI,
                                                  const float* __restrict__ xl,
                                                  const float* __restrict__ xr,
                                                  float* __restrict__ lg,
                                                  float* __restrict__ amax) {
    const int lane = threadIdx.x;
    const int ln   = lane & 15;
    const int half = lane >> 4;
    const int e0   = blockIdx.x * 16;

    // Preload A (edge_attr rows, K=64): 16 k-steps x 2 floats per lane.
    const float* arow = ea + (size_t)(e0 + ln) * EDIM + half * 2;
    float a[32];
#pragma unroll
    for (int kk = 0; kk < 16; ++kk) {
        a[2 * kk]     = arow[4 * kk];
        a[2 * kk + 1] = arow[4 * kk + 1];
    }

    // Edge indices for the rows this VGPR-slot owns (D layout rows).
    int sv[8], dv[8];
#pragma unroll
    for (int v = 0; v < 8; ++v) {
        const int e = e0 + v + 8 * half;
        sv[v] = srcI[e];
        dv[v] = dstI[e];
    }

    float acc[8][4];
#pragma unroll
    for (int v = 0; v < 8; ++v)
#pragma unroll
        for (int h = 0; h < 4; ++h) acc[v][h] = 0.0f;

#pragma unroll
    for (int nt = 0; nt < 8; ++nt) {         // 8 column tiles of 16 channels
        const float* bcol = We + nt * 16 + ln;
        v8f c = {};
#pragma unroll
        for (int kk = 0; kk < 16; ++kk) {    // K = 64
            v2f av, bv;
            av.x = a[2 * kk];
            av.y = a[2 * kk + 1];
            bv.x = bcol[(size_t)(kk * 4 + half * 2) * HC];
            bv.y = bcol[(size_t)(kk * 4 + half * 2 + 1) * HC];
            c = __builtin_amdgcn_wmma_f32_16x16x4_f32(false, av, false, bv,
                                                      (short)0, c, false, false);
        }
        const int   cl   = nt * 16 + ln;     // global channel 0..127
        const float attv = att[cl];          // att flat [H*C] == channel index
        const int   h    = nt >> 1;          // head of this column tile
#pragma unroll
        for (int v = 0; v < 8; ++v) {
            float mv = c[v] + xl[(size_t)sv[v] * HC + cl]
                            + xr[(size_t)dv[v] * HC + cl];
            mv = mv > 0.0f ? mv : 0.2f * mv; // LeakyReLU(0.2)
            acc[v][h] += mv * attv;
        }
    }

    // Reduce channel partial sums across each 16-lane half.
#pragma unroll
    for (int v = 0; v < 8; ++v)
#pragma unroll
        for (int h = 0; h < 4; ++h) {
            float s = acc[v][h];
            s += __shfl_xor(s, 8, 16);
            s += __shfl_xor(s, 4, 16);
            s += __shfl_xor(s, 2, 16);
            s += __shfl_xor(s, 1, 16);
            acc[v][h] = s;
        }

    if (ln == 0) {
#pragma unroll
        for (int v = 0; v < 8; ++v) {
            const int e = e0 + v + 8 * half;
#pragma unroll
            for (int h = 0; h < 4; ++h) {
                const float L = acc[v][h];
                lg[(size_t)e * HH + h] = L;
                atomicMaxF(&amax[(size_t)dv[v] * HH + h], L);
            }
        }
    }
}

// ---------------------------------------------------------------------------
// ex = exp(logit - amax[dst]) (in-place over logits) + denom segment sum.
// ---------------------------------------------------------------------------
__global__ void __launch_bounds__(256) exp_denom(const int* __restrict__ dstI,
                                                 const float* __restrict__ amax,
                                                 float* __restrict__ lg,
                                                 float* __restrict__ denom) {
    const int t = blockIdx.x * blockDim.x + threadIdx.x;
    if (t >= EE * HH) return;
    const int e = t >> 2;
    const int h = t & 3;
    const int d = dstI[e];
    const float ex = expf(lg[t] - amax[(size_t)d * HH + h]);
    lg[t] = ex;
    atomicAdd(&denom[(size_t)d * HH + h], ex);
}

// ---------------------------------------------------------------------------
// out[dst] += alpha * x_l[src]; one block per edge, one thread per channel.
// ---------------------------------------------------------------------------
__global__ void __launch_bounds__(128) aggregate(const int* __restrict__ srcI,
                                                 const int* __restrict__ dstI,
                                                 const float* __restrict__ xl,
                                                 const float* __restrict__ lg,
                                                 const float* __restrict__ denom,
                                                 float* __restrict__ out) {
    const int e = blockIdx.x;
    const int c = threadIdx.x;
    const int h = c >> 5;
    const int s = srcI[e];
    const int d = dstI[e];
    const float alpha = lg[(size_t)e * HH + h] /
                        (denom[(size_t)d * HH + h] + 1e-16f);
    atomicAdd(&out[(size_t)d * HC + c], alpha * xl[(size_t)s * HC + c]);
}

// ---------------------------------------------------------------------------
// out = bias (accumulator init), amax = -inf, denom = 0.
// ---------------------------------------------------------------------------
__global__ void __launch_bounds__(256) init_ws(float* __restrict__ out,
                                               const float* __restrict__ bias,
                                               float* __restrict__ amax,
                                               float* __restrict__ denom) {
    const int i = blockIdx.x * blockDim.x + threadIdx.x;
    if (i < NN * HC) out[i] = bias[i & (HC - 1)];
    if (i < NN * HH) {
        amax[i]  = __int_as_float(0xff800000); // -inf
        denom[i] = 0.0f;
    }
}

extern "C" void kernel_launch(void* const* d_in, const int* in_sizes, int n_in,
                              void* d_out, int out_size, void* d_ws, size_t ws_size,
                              hipStream_t stream) {
    (void)in_sizes; (void)n_in; (void)out_size; (void)ws_size;
    const float* x    = (const float*)d_in[0];
    const float* ea   = (const float*)d_in[1];
    const float* Wl   = (const float*)d_in[2];
    const float* Wr   = (const float*)d_in[3];
    const float* We   = (const float*)d_in[4];
    const float* att  = (const float*)d_in[5];
    const float* bias = (const float*)d_in[6];
    const int*   ei   = (const int*)d_in[7];   // [2,E]: row0=src, row1=dst
    const int* srcI = ei;
    const int* dstI = ei + EE;
    float* out = (float*)d_out;

    float* ws    = (float*)d_ws;
    float* xl    = ws;                                 // N*HC
    float* xr    = xl + (size_t)NN * HC;               // N*HC
    float* lg    = xr + (size_t)NN * HC;               // E*H (logits -> ex)
    float* amax  = lg + (size_t)EE * HH;               // N*H
    float* denom = amax + (size_t)NN * HH;             // N*H

    init_ws<<<(NN * HC + 255) / 256, 256, 0, stream>>>(out, bias, amax, denom);
    node_gemm<<<dim3(NN / 16, HC / 16, 2), 32, 0, stream>>>(x, Wl, Wr, xl, xr);
    edge_logits<<<EE / 16, 32, 0, stream>>>(ea, We, att, srcI, dstI, xl, xr, lg, amax);
    exp_denom<<<(EE * HH + 255) / 256, 256, 0, stream>>>(dstI, amax, lg, denom);
    aggregate<<<EE, 128, 0, stream>>>(srcI, dstI, xl, lg, denom, out);
}